// GAE_3796751090175
// MI455X (gfx1250) — compile-verified
//
#include <hip/hip_runtime.h>
#include <math.h>

// ---- problem constants (from reference) ----
#define NUSERS 6040
#define NITEMS 3706
#define RCLS   5
#define IND    128
#define H0     64
#define H1     32
#define NBB    2
#define NU     2048
#define NV     2048
#define NN     4096   // NU + NV
#define KCHUNK 512    // K split for the graph-conv WMMA kernel (4 chunks of 512)

// ---- WMMA vector types (f32 16x16x4: A/B = 2 VGPRs, C/D = 8 VGPRs) ----
typedef __attribute__((ext_vector_type(2))) float v2f;
typedef __attribute__((ext_vector_type(8))) float v8f;

// ---- workspace layout (float offsets), total ~2.04M floats (~8.2 MB) ----
#define WS_DEG   0                              // 4096: [du(2048) | di(2048)] -> c in place
#define WS_SUP   4096                           // RCLS*NN*H0 = 1,310,720
#define WS_HID   (WS_SUP + RCLS*NN*H0)          // NN*H1 = 131,072
#define WS_RUE   (WS_HID + NN*H1)               // RCLS*NU*H1 = 327,680
#define WS_GOUT  (WS_RUE + RCLS*NU*H1)          // NN*H0 = 262,144 (conv accumulator)
#define WS_SCAL  (WS_GOUT + NN*H0)              // 3: [total_obs, loss_acc, rmse_acc]

static __device__ __forceinline__ float wave_reduce_add(float x) {
  for (int off = 16; off > 0; off >>= 1) x += __shfl_down(x, off);   // wave32
  return x;
}

// ---------------- K0: zero per-call accumulators (gout + scalars) ----------------
__global__ void k_zero(float* __restrict__ ws) {
  const int i = blockIdx.x * 256 + threadIdx.x;
  if (i < NN * H0 + 3) ws[WS_GOUT + i] = 0.0f;
}

// ---------------- K1a: du[j] = |sum_{r,i} m[r,u_i,v_j]| ----------------
__global__ void k_du(const float* __restrict__ m, const int* __restrict__ u,
                     const int* __restrict__ v, float* __restrict__ ws) {
  __shared__ float red[256];
  const int j  = blockIdx.x;
  const int vj = v[j];
  float s = 0.0f;
  for (int r = 0; r < RCLS; ++r) {
    const float* base = m + (size_t)r * NUSERS * NITEMS + vj;
    for (int i = threadIdx.x; i < NU; i += 256)
      s += base[(size_t)u[i] * NITEMS];
  }
  red[threadIdx.x] = s;
  __syncthreads();
  for (int st = 128; st > 0; st >>= 1) {
    if (threadIdx.x < st) red[threadIdx.x] += red[threadIdx.x + st];
    __syncthreads();
  }
  if (threadIdx.x == 0) ws[WS_DEG + j] = fabsf(red[0]);
}

// ---------------- K1b: di[i] = |sum_{r,j} m[r,u_i,v_j]|, also total obs ----------------
__global__ void k_di(const float* __restrict__ m, const int* __restrict__ u,
                     const int* __restrict__ v, float* __restrict__ ws) {
  __shared__ float red[256];
  const int i  = blockIdx.x;
  const int ui = u[i];
  float s = 0.0f;
  for (int r = 0; r < RCLS; ++r) {
    const float* row = m + ((size_t)r * NUSERS + ui) * NITEMS;
    for (int j = threadIdx.x; j < NV; j += 256)
      s += row[v[j]];
  }
  red[threadIdx.x] = s;
  __syncthreads();
  for (int st = 128; st > 0; st >>= 1) {
    if (threadIdx.x < st) red[threadIdx.x] += red[threadIdx.x + st];
    __syncthreads();
  }
  if (threadIdx.x == 0) {
    ws[WS_DEG + NU + i] = fabsf(red[0]);
    atomicAdd(&ws[WS_SCAL + 0], red[0]);   // total observation count (m >= 0)
  }
}

// ---------------- K1c: c[n] = deg[n] > 0 ? 1/deg[n] : 0 (in place) ----------------
__global__ void k_recip(float* __restrict__ ws) {
  const int n = blockIdx.x * blockDim.x + threadIdx.x;
  if (n < NN) {
    const float d = ws[WS_DEG + n];
    ws[WS_DEG + n] = (d > 0.0f) ? (1.0f / d) : 0.0f;
  }
}

// ---------------- K2: sup[r,n,h] = x[n,:] @ gcl_w[r]  (WMMA f32 16x16x4) ----------------
// grid = (NN/16, H0/16, RCLS), block = 32 (one wave, EXEC all ones)
__global__ __launch_bounds__(32) void k_sup(
    const int* __restrict__ u, const int* __restrict__ v,
    const float* __restrict__ u_emb, const float* __restrict__ v_emb,
    const float* __restrict__ gcl_w, float* __restrict__ ws) {
  const int lane = threadIdx.x;
  const int mt = blockIdx.x * 16;
  const int nt = blockIdx.y * 16;
  const int r  = blockIdx.z;

  // A layout (16x4 f32): lanes 0-15 -> row M=lane&15, K pair {0,1}; lanes 16-31 -> K pair {2,3}
  const int arow = mt + (lane & 15);
  const int koff = (lane < 16) ? 0 : 2;
  const float* xrow = (arow < NU) ? (u_emb + (size_t)u[arow] * IND)
                                  : (v_emb + (size_t)v[arow - NU] * IND);
  const int bcol = nt + (lane & 15);
  const float* W = gcl_w + (size_t)r * IND * H0;

  v8f acc = {0.f, 0.f, 0.f, 0.f, 0.f, 0.f, 0.f, 0.f};
  for (int k = 0; k < IND; k += 4) {
    v2f av, bv;
    av.x = xrow[k + koff];
    av.y = xrow[k + koff + 1];
    bv.x = W[(size_t)(k + koff) * H0 + bcol];
    bv.y = W[(size_t)(k + koff + 1) * H0 + bcol];
    acc = __builtin_amdgcn_wmma_f32_16x16x4_f32(false, av, false, bv,
                                                (short)0, acc, false, false);
  }
  float* supr = ws + WS_SUP + (size_t)r * NN * H0;
  const int rbase = mt + ((lane < 16) ? 0 : 8);
  #pragma unroll
  for (int i = 0; i < 8; ++i)
    supr[(size_t)(rbase + i) * H0 + (nt + (lane & 15))] = acc[i];
}

// ---------------- K3: graph conv as WMMA GEMM with gathered A-matrix ----------------
// out_u[i,h] = sum_{r,j} m[r,u_i,v_j] * sup[r,NU+j,h]   (tiles mt <  NU)
// out_v[j,h] = sum_{r,i} m[r,u_i,v_j] * sup[r,   i,h]   (tiles mt >= NU)
// grid = (NN/16, H0/16, RCLS*4): z = r*4 + K-chunk; partials atomically reduced into gout.
__global__ __launch_bounds__(32) void k_gconv(
    const float* __restrict__ m, const int* __restrict__ u, const int* __restrict__ v,
    float* __restrict__ ws) {
  __shared__ int idx[KCHUNK];
  const int lane = threadIdx.x;
  const int mt = blockIdx.x * 16;
  const int nt = blockIdx.y * 16;
  const int r  = blockIdx.z >> 2;
  const int k0 = (blockIdx.z & 3) * KCHUNK;
  const bool uside = (mt < NU);            // uniform per wave (NU % 16 == 0)

  // stage this chunk's gather indices in LDS
  for (int t = lane; t < KCHUNK; t += 32)
    idx[t] = uside ? v[k0 + t] : u[k0 + t];
  __syncthreads();

  const int koff = (lane < 16) ? 0 : 2;
  const int arow = mt + (lane & 15);
  const int bcol = nt + (lane & 15);
  // A-element gather base for this lane's fixed row of m
  const float* mrbase = uside
      ? (m + ((size_t)r * NUSERS + u[arow]) * NITEMS)          // + v[k]
      : (m + (size_t)r * NUSERS * NITEMS + v[arow - NU]);      // + u[k]*NITEMS
  const float* supr = ws + WS_SUP + ((size_t)r * NN + (uside ? NU : 0) + k0) * H0;

  v8f acc = {0.f, 0.f, 0.f, 0.f, 0.f, 0.f, 0.f, 0.f};
  if (uside) {
    for (int k = 0; k < KCHUNK; k += 4) {
      v2f av, bv;
      av.x = mrbase[idx[k + koff]];
      av.y = mrbase[idx[k + koff + 1]];
      bv.x = supr[(size_t)(k + koff) * H0 + bcol];
      bv.y = supr[(size_t)(k + koff + 1) * H0 + bcol];
      acc = __builtin_amdgcn_wmma_f32_16x16x4_f32(false, av, false, bv,
                                                  (short)0, acc, false, false);
    }
  } else {
    for (int k = 0; k < KCHUNK; k += 4) {
      v2f av, bv;
      av.x = mrbase[(size_t)idx[k + koff] * NITEMS];
      av.y = mrbase[(size_t)idx[k + koff + 1] * NITEMS];
      bv.x = supr[(size_t)(k + koff) * H0 + bcol];
      bv.y = supr[(size_t)(k + koff + 1) * H0 + bcol];
      acc = __builtin_amdgcn_wmma_f32_16x16x4_f32(false, av, false, bv,
                                                  (short)0, acc, false, false);
    }
  }

  float* go = ws + WS_GOUT;
  const int rbase = mt + ((lane < 16) ? 0 : 8);
  #pragma unroll
  for (int i = 0; i < 8; ++i)
    atomicAdd(&go[(size_t)(rbase + i) * H0 + (nt + (lane & 15))], acc[i]);
}

// ---------------- K4: z = gout*c + 5*b, relu, dense, sigmoid -> hidden ----------------
// grid = NN blocks, block = 64 threads (thread = h)
__global__ __launch_bounds__(64) void k_dense(
    const float* __restrict__ gcl_b, const float* __restrict__ dense_w,
    const float* __restrict__ dense_b, float* __restrict__ ws) {
  __shared__ float zs[H0];
  const int n = blockIdx.x;
  const int h = threadIdx.x;
  const float z = ws[WS_GOUT + (size_t)n * H0 + h] * ws[WS_DEG + n]
                + (float)RCLS * gcl_b[h];
  zs[h] = fmaxf(z, 0.0f);
  __syncthreads();
  if (h < H1) {
    float s = dense_b[h];
    for (int k = 0; k < H0; ++k) s += zs[k] * dense_w[k * H1 + h];
    ws[WS_HID + (size_t)n * H1 + h] = 1.0f / (1.0f + expf(-s));
  }
}

// ---------------- K5: rue[r,i,e] = sum_d hu[i,d] * (a[r,0]P0 + a[r,1]P1)[d,e] ----------------
__global__ __launch_bounds__(32) void k_rue(
    const float* __restrict__ P, const float* __restrict__ a, float* __restrict__ ws) {
  __shared__ float hs[H1];
  const int i = blockIdx.x;
  const int e = threadIdx.x;
  hs[e] = ws[WS_HID + (size_t)i * H1 + e];
  __syncthreads();
  for (int r = 0; r < RCLS; ++r) {
    const float a0 = a[r * NBB + 0], a1 = a[r * NBB + 1];
    float s = 0.0f;
    for (int d = 0; d < H1; ++d) {
      const float q = a0 * P[d * H1 + e] + a1 * P[H1 * H1 + d * H1 + e];
      s += hs[d] * q;
    }
    ws[WS_RUE + ((size_t)r * NU + i) * H1 + e] = s;
  }
}

// ---------------- K6: fused logits (WMMA) + softmax(r) + m_hat + loss/rmse ----------------
// grid = (NU/16, NV/16), block = 32. Never materializes logits in HBM.
__global__ __launch_bounds__(32) void k_decode(
    const float* __restrict__ m, const int* __restrict__ u, const int* __restrict__ v,
    float* __restrict__ out, float* __restrict__ ws) {
  const int lane = threadIdx.x;
  const int ut = blockIdx.x * 16;
  const int vt = blockIdx.y * 16;
  const int koff = (lane < 16) ? 0 : 2;
  const int arow = ut + (lane & 15);
  const int bcol = vt + (lane & 15);
  const float* hvrow = ws + WS_HID + (size_t)(NU + bcol) * H1;   // B[k=e,n=j] = hv[j,e]

  v8f acc[RCLS];
  #pragma unroll
  for (int r = 0; r < RCLS; ++r) {
    v8f c = {0.f, 0.f, 0.f, 0.f, 0.f, 0.f, 0.f, 0.f};
    const float* ar = ws + WS_RUE + ((size_t)r * NU + arow) * H1;
    for (int e = 0; e < H1; e += 4) {
      v2f av, bv;
      av.x = ar[e + koff];
      av.y = ar[e + koff + 1];
      bv.x = hvrow[e + koff];
      bv.y = hvrow[e + koff + 1];
      c = __builtin_amdgcn_wmma_f32_16x16x4_f32(false, av, false, bv,
                                                (short)0, c, false, false);
    }
    acc[r] = c;
  }

  const int rbase = ut + ((lane < 16) ? 0 : 8);
  const int col   = vt + (lane & 15);
  const int vj    = v[col];
  float loss_local = 0.0f, rmse_local = 0.0f;

  #pragma unroll
  for (int i = 0; i < 8; ++i) {
    const int row = rbase + i;
    float lg[RCLS];
    float mx = -1e30f;
    #pragma unroll
    for (int r = 0; r < RCLS; ++r) { lg[r] = acc[r][i]; mx = fmaxf(mx, lg[r]); }
    float se = 0.0f;
    float ex[RCLS];
    #pragma unroll
    for (int r = 0; r < RCLS; ++r) { ex[r] = expf(lg[r] - mx); se += ex[r]; }
    const float inv = 1.0f / se;
    const float lse = mx + logf(se);
    float mhat = 0.0f;
    #pragma unroll
    for (int r = 0; r < RCLS; ++r) mhat += (float)(r + 1) * ex[r] * inv;

    const int ui = u[row];
    float obs = 0.0f, rating = 0.0f;
    #pragma unroll
    for (int r = 0; r < RCLS; ++r) {
      const float mv = m[((size_t)r * NUSERS + ui) * NITEMS + vj];
      obs    += mv;
      rating += (float)(r + 1) * mv;
      loss_local -= mv * (lg[r] - lse);
    }
    const float d = (mhat - rating) * obs;
    rmse_local += d * d;
    out[(size_t)row * NV + col] = mhat;
  }

  loss_local = wave_reduce_add(loss_local);
  rmse_local = wave_reduce_add(rmse_local);
  if (lane == 0) {
    atomicAdd(&ws[WS_SCAL + 1], loss_local);
    atomicAdd(&ws[WS_SCAL + 2], rmse_local);
  }
}

// ---------------- K7: finalize scalars ----------------
__global__ void k_final(float* __restrict__ out, const float* __restrict__ ws) {
  if (threadIdx.x == 0 && blockIdx.x == 0) {
    const float cnt = fmaxf(ws[WS_SCAL + 0], 1.0f);
    out[(size_t)NU * NV + 0] = ws[WS_SCAL + 1] / cnt;
    out[(size_t)NU * NV + 1] = sqrtf(ws[WS_SCAL + 2] / cnt);
  }
}

extern "C" void kernel_launch(void* const* d_in, const int* in_sizes, int n_in,
                              void* d_out, int out_size, void* d_ws, size_t ws_size,
                              hipStream_t stream) {
  const int*   u       = (const int*)  d_in[0];
  const int*   v       = (const int*)  d_in[1];
  const float* m       = (const float*)d_in[2];
  const float* u_emb   = (const float*)d_in[3];
  const float* v_emb   = (const float*)d_in[4];
  const float* gcl_w   = (const float*)d_in[5];
  const float* gcl_b   = (const float*)d_in[6];
  const float* dense_w = (const float*)d_in[7];
  const float* dense_b = (const float*)d_in[8];
  const float* P       = (const float*)d_in[9];
  const float* a       = (const float*)d_in[10];
  float*       out     = (float*)d_out;
  float*       ws      = (float*)d_ws;
  (void)in_sizes; (void)n_in; (void)out_size; (void)ws_size;

  k_zero <<<(NN * H0 + 3 + 255) / 256, 256, 0, stream>>>(ws);
  k_du   <<<NV, 256, 0, stream>>>(m, u, v, ws);
  k_di   <<<NU, 256, 0, stream>>>(m, u, v, ws);
  k_recip<<<(NN + 255) / 256, 256, 0, stream>>>(ws);
  k_sup  <<<dim3(NN / 16, H0 / 16, RCLS), 32, 0, stream>>>(u, v, u_emb, v_emb, gcl_w, ws);
  k_gconv<<<dim3(NN / 16, H0 / 16, RCLS * 4), 32, 0, stream>>>(m, u, v, ws);
  k_dense<<<NN, 64, 0, stream>>>(gcl_b, dense_w, dense_b, ws);
  k_rue  <<<NU, 32, 0, stream>>>(P, a, ws);
  k_decode<<<dim3(NU / 16, NV / 16), 32, 0, stream>>>(m, u, v, out, ws);
  k_final<<<1, 32, 0, stream>>>(out, ws);
}